// EntityEncoder_28845000360091
// MI455X (gfx1250) — compile-verified
//
#include <hip/hip_runtime.h>
#include <hip/hip_bf16.h>

// Problem constants (from reference)
#define B_  256
#define N_  4096
#define F_  8192
#define NT_ 512
#define TD_ 122                       // TYPE_DIM = 128 - 6
#define EMB_FLOATS (NT_ * TD_)        // 62464 floats = 249856 bytes (fits 320KB LDS)
#define EMB_VEC4   (EMB_FLOATS / 4)   // 15616 float4 chunks

typedef __attribute__((ext_vector_type(4))) float    v4f;
typedef __attribute__((ext_vector_type(4))) int      v4i;
typedef __attribute__((ext_vector_type(4))) unsigned u32x4;
typedef __attribute__((ext_vector_type(8))) int      i32x8;
typedef __attribute__((ext_vector_type(4))) int      i32x4;

#define AS_GLOBAL __attribute__((address_space(1)))
#define AS_LDS    __attribute__((address_space(3)))

#if defined(__has_builtin)
#if __has_builtin(__builtin_amdgcn_tensor_load_to_lds)
#define USE_TDM 1
#endif
#if __has_builtin(__builtin_amdgcn_global_load_async_to_lds_b128)
#define HAVE_ASYNC_LDS 1
#endif
#endif

// ---------------- kernel 1: zero the counts scratch (B*N uint32) ----------------
__global__ void ee_zero_counts(unsigned* __restrict__ ws) {
  int i = blockIdx.x * blockDim.x + threadIdx.x;
  if (i < (B_ * N_) / 4) {
    uint4 z; z.x = 0u; z.y = 0u; z.z = 0u; z.w = 0u;
    ((uint4*)ws)[i] = z;
  }
}

// ---------------- kernel 2: histogram of facts[:,:,1] ----------------
__global__ void ee_hist(const int2* __restrict__ facts, unsigned* __restrict__ counts) {
  int i = blockIdx.x * blockDim.x + threadIdx.x;     // one fact per thread
  if (i < B_ * F_) {
    int2 p = facts[i];                               // coalesced b64 load
    int b = i >> 13;                                 // i / F_
    atomicAdd(&counts[b * N_ + p.y], 1u);
  }
}

// ---------------- kernel 3: encode ----------------
// One wave per entity row: 32 lanes x float4 = 128 output floats.
// Embedding table staged to LDS by the Tensor Data Mover (one DMA per block).
__global__ __launch_bounds__(1024) void ee_encode(
    const float* __restrict__ entities,
    const unsigned* __restrict__ counts,
    const float* __restrict__ type_emb,
    float* __restrict__ out) {
  __shared__ float semb[EMB_FLOATS];                 // 249856 B static LDS
  const int tid = threadIdx.x;

#if defined(USE_TDM)
  if (tid < 32) {  // wave 0 only: TDM ignores EXEC; one DMA per workgroup
    unsigned long long ga = (unsigned long long)(size_t)type_emb;
    unsigned lds_off = (unsigned)(size_t)(AS_LDS char*)semb;  // AS3 repr == LDS byte offset

    // D# group 0: count=1 | lds_addr | global_addr[56:0] | type=2
    u32x4 g0;
    g0.x = 1u;                                            // count=1, user mode, no gather
    g0.y = lds_off;                                       // lds_addr (bytes)
    g0.z = (unsigned)ga;                                  // global_addr[31:0]
    g0.w = (unsigned)((ga >> 32) & 0x01FFFFFFu) | (2u << 30);  // addr[56:32] | type=2

    // D# group 1: 1-D tile of 62464 x 4B elements, no multicast/pad/barrier
    i32x8 g1;
    g1[0] = (int)(2u << 16);             // workgroup_mask=0, data_size=2 (4 bytes)
    g1[1] = (int)((unsigned)EMB_FLOATS << 16);  // tensor_dim0[15:0]=62464 (atomic_barrier_addr=0)
    g1[2] = (int)(1u << 16);             // tensor_dim0[31:16]=0, tensor_dim1[15:0]=1
    g1[3] = (int)((unsigned)EMB_FLOATS << 16);  // tensor_dim1[31:16]=0, tile_dim0=62464
    g1[4] = 1;                           // tile_dim1=1, tile_dim2=0
    g1[5] = EMB_FLOATS;                  // tensor_dim0_stride[31:0]
    g1[6] = 0;                           // stride0[47:32]=0, stride1[15:0]=0
    g1[7] = 0;                           // stride1[47:16]=0

    i32x4 gz4 = {0, 0, 0, 0};            // groups 2/3 unused (<=2D tensor)
#if __clang_major__ >= 23
    i32x8 gz8 = {0, 0, 0, 0, 0, 0, 0, 0};
    __builtin_amdgcn_tensor_load_to_lds(g0, g1, gz4, gz4, gz8, 0);
#else
    __builtin_amdgcn_tensor_load_to_lds(g0, g1, gz4, gz4, 0);
#endif
#if __has_builtin(__builtin_amdgcn_s_wait_tensorcnt)
    __builtin_amdgcn_s_wait_tensorcnt(0);
#else
    asm volatile("s_wait_tensorcnt 0" ::: "memory");
#endif
  }
#elif defined(HAVE_ASYNC_LDS)
  {
    const AS_GLOBAL char* g = (const AS_GLOBAL char*)type_emb;
    AS_LDS char* l = (AS_LDS char*)semb;
    for (int i = tid; i < EMB_VEC4; i += 1024) {
      __builtin_amdgcn_global_load_async_to_lds_b128(
          (AS_GLOBAL v4i*)(g + (size_t)i * 16),
          (AS_LDS v4i*)(l + (size_t)i * 16),
          0, 0);
    }
#if __has_builtin(__builtin_amdgcn_s_wait_asynccnt)
    __builtin_amdgcn_s_wait_asynccnt(0);
#else
    asm volatile("s_wait_asynccnt 0" ::: "memory");
#endif
  }
#else
  for (int i = tid; i < EMB_VEC4; i += 1024)
    ((v4f*)semb)[i] = ((const v4f*)type_emb)[i];
#endif
  __syncthreads();

  const int lane  = tid & 31;
  const int wave  = tid >> 5;
  int gwave  = blockIdx.x * 32 + wave;               // 32 waves per block
  int nwaves = gridDim.x * 32;

  for (int e = gwave; e < B_ * N_; e += nwaves) {
    const float* ent = entities + (size_t)e * 5;
    float e1 = ent[1];
    float az = ent[2];
    float e3 = ent[3];
    int   t  = (int)ent[4];

    float dn = fabsf(az) * (1.0f / 180.0f);
    float de = (az >= -90.0f ? fabsf(90.0f - az)
                             : 90.0f + fabsf(az + 180.0f)) * (1.0f / 180.0f);
    float cnt = ((e & (N_ - 1)) == (N_ - 1)) ? 0.0f : (float)counts[e];  // zero col N-1
    float ind = (cnt > 0.0f) ? 1.0f : 0.0f;

    const float* erow = semb + t * TD_;              // LDS gather, wave-uniform row
    v4f v;
    if (lane == 0) {
      v.x = e1;  v.y = dn;  v.z = de;      v.w = e3;
    } else if (lane == 1) {
      v.x = cnt; v.y = ind; v.z = erow[0]; v.w = erow[1];
    } else {
      int k = lane * 4 - 6;                          // channels 8..127 -> emb[2..121]
      v.x = erow[k]; v.y = erow[k + 1]; v.z = erow[k + 2]; v.w = erow[k + 3];
    }
    // streaming NT store: 512B per wave instruction, fully coalesced
    __builtin_nontemporal_store(v, ((v4f*)out) + (size_t)e * 32 + lane);
  }
}

extern "C" void kernel_launch(void* const* d_in, const int* in_sizes, int n_in,
                              void* d_out, int out_size, void* d_ws, size_t ws_size,
                              hipStream_t stream) {
  (void)in_sizes; (void)n_in; (void)out_size; (void)ws_size;
  const float* entities  = (const float*)d_in[0];    // (B,N,5) f32
  const int*   facts     = (const int*)d_in[1];      // (B,F,2) i32
  const float* type_emb  = (const float*)d_in[2];    // (NUM_TYPES,TYPE_DIM) f32
  float*       out       = (float*)d_out;            // (B,N,128) f32
  unsigned*    counts    = (unsigned*)d_ws;          // B*N u32 = 4 MiB scratch

  // 1) zero counts (scratch is poisoned; must re-zero every call)
  ee_zero_counts<<<(B_ * N_ / 4 + 255) / 256, 256, 0, stream>>>(counts);
  // 2) histogram over facts[:,:,1]
  ee_hist<<<(B_ * F_ + 255) / 256, 256, 0, stream>>>((const int2*)facts, counts);
  // 3) fused encode + embedding gather (one wave per output row)
  ee_encode<<<512, 1024, 0, stream>>>(entities, counts, type_emb, out);
}